// MDNMV_59356448030796
// MI455X (gfx1250) — compile-verified
//
#include <hip/hip_runtime.h>

// ---------------------------------------------------------------------------
// MDN multivariate-normal log-likelihood. B=16384 rows, K=32 components,
// D=16, 153 floats/component. Bandwidth bound: 321 MB @ 23.3 TB/s ~ 13.8 us.
// wave32 = one batch row (lane = component). Row staging via TDM
// TENSOR_LOAD_TO_LDS (one DMA descriptor per wave; async-b128 fallback),
// conflict-free LDS reads (stride 153 coprime to 64 banks), fully unrolled
// register forward substitution with raw v_exp/v_log softplus, and both K=32
// logsumexp reductions folded into one v_wmma_f32_16x16x4_f32 per row.
// ---------------------------------------------------------------------------

typedef __attribute__((ext_vector_type(2))) float v2f;
typedef __attribute__((ext_vector_type(8))) float v8f;
typedef int v4i __attribute__((vector_size(4 * sizeof(int))));
typedef unsigned int u32x4 __attribute__((vector_size(16)));
typedef int i32x4 __attribute__((vector_size(16)));
typedef int i32x8 __attribute__((vector_size(32)));

#define KCOMP 32
#define DDIM 16
#define NPC 153               // 1 + 16 + 136 floats per component
#define ROWF (KCOMP * NPC)    // 4896 floats per batch row
#define ROWS 4                // batch rows per block (1 per wave)
#define BLOCK (ROWS * 32)
#define LOG_2PI 1.8378770664093453f
#define LOG2E 1.4426950408889634f
#define LN2 0.6931471805599453f

#ifndef __has_builtin
#define __has_builtin(x) 0
#endif

#if __has_builtin(__builtin_amdgcn_global_load_async_to_lds_b128) && \
    __has_builtin(__builtin_amdgcn_global_load_async_to_lds_b32) &&  \
    __has_builtin(__builtin_amdgcn_s_wait_asynccnt)
#define USE_ASYNC_LDS 1
#else
#define USE_ASYNC_LDS 0
#endif

#if __has_builtin(__builtin_amdgcn_tensor_load_to_lds) && \
    __has_builtin(__builtin_amdgcn_s_wait_tensorcnt)
#define USE_TDM 1
#else
#define USE_TDM 0
#endif

#if __has_builtin(__builtin_amdgcn_exp2f) && __has_builtin(__builtin_amdgcn_logf)
#define FAST_EXP2(x) __builtin_amdgcn_exp2f(x)     // v_exp_f32 (base 2)
#define FAST_LOG2(x) __builtin_amdgcn_logf(x)      // v_log_f32 (base 2)
#else
#define FAST_EXP2(x) __expf((x) * LN2)
#define FAST_LOG2(x) (__logf(x) * LOG2E)
#endif

#define GAS128(p) ((__attribute__((address_space(1))) v4i*)(p))
#define LAS128(p) ((__attribute__((address_space(3))) v4i*)(p))
#define GAS32(p)  ((__attribute__((address_space(1))) int*)(p))
#define LAS32(p)  ((__attribute__((address_space(3))) int*)(p))

__global__ __launch_bounds__(BLOCK) void mdn_mv_logp_kernel(
    const float* __restrict__ p, const float* __restrict__ y,
    float* __restrict__ out, int nrows) {
  extern __shared__ float lds[];            // ROWS*ROWF + ROWS*DDIM floats
  float* prow = lds;
  float* yrow = lds + ROWS * ROWF;

  const int lane = threadIdx.x & 31;
  const int wave = threadIdx.x >> 5;
  const int b = blockIdx.x * ROWS + wave;
  const int brow = (b < nrows) ? b : (nrows - 1);  // clamp: keep waves full

  // ---- stage this wave's row of p (4896 floats = one 1-D tile) into LDS --
  const float* gsrc = p + (size_t)brow * ROWF;
  float* ldst = prow + wave * ROWF;

#if USE_TDM
  {
    // Tensor DMA descriptor (ISA 8.3/8.4): 1-D tile, 4-byte elements,
    // tensor_dim0 = tile_dim0 = stride0 = 4896, dim1 = tile_dim1 = 1.
    const unsigned long long ga = (unsigned long long)(uintptr_t)gsrc;
    const unsigned la =
        (unsigned)(uintptr_t)(__attribute__((address_space(3))) float*)ldst;
    u32x4 g0;
    g0[0] = 1u;                                   // count=1, user descriptor
    g0[1] = la;                                   // lds_addr (bytes)
    g0[2] = (unsigned)(ga & 0xffffffffu);         // global_addr[31:0]
    g0[3] = (unsigned)((ga >> 32) & 0x01ffffffu)  // global_addr[56:32]
            | (2u << 30);                         // type = 2 ("image")
    i32x8 g1;
    g1[0] = (int)(2u << 16);                      // data_size=2 (4B), mask=0
    g1[1] = (int)((ROWF & 0xffffu) << 16);        // tensor_dim0[15:0] @ 63:48
    g1[2] = (int)(((unsigned)ROWF >> 16) | (1u << 16)); // dim0 hi, dim1=1
    g1[3] = (int)((ROWF & 0xffffu) << 16);        // tile_dim0 @ 127:112
    g1[4] = 1;                                    // tile_dim1=1, tile_dim2=0
    g1[5] = (int)ROWF;                            // dim0_stride[31:0]
    g1[6] = (int)((ROWF & 0xffffu) << 16);        // stride0 hi=0 | stride1 lo
    g1[7] = 0;                                    // stride1 hi
    const i32x4 gz4 = {0, 0, 0, 0};               // groups 2/3 unused (<=2D)
    const i32x8 gz8 = {0, 0, 0, 0, 0, 0, 0, 0};
    __builtin_amdgcn_tensor_load_to_lds(g0, g1, gz4, gz4, gz8, 0);
  }
#elif USE_ASYNC_LDS
  #pragma unroll
  for (int i = 0; i < 38; ++i) {
    __builtin_amdgcn_global_load_async_to_lds_b128(
        GAS128(gsrc + (lane + i * 32) * 4),
        LAS128(ldst + (lane + i * 32) * 4), 0, 0);
  }
  __builtin_amdgcn_global_load_async_to_lds_b32(
      GAS32(gsrc + 4864 + lane), LAS32(ldst + 4864 + lane), 0, 0);
#else
  {
    const float4* g4 = (const float4*)gsrc;
    float4* l4 = (float4*)ldst;
    #pragma unroll
    for (int i = 0; i < 38; ++i) l4[lane + i * 32] = g4[lane + i * 32];
    ldst[4864 + lane] = gsrc[4864 + lane];
  }
#endif

  // ---- y row (16 floats) --------------------------------------------------
#if USE_ASYNC_LDS
  if (lane < DDIM) {
    __builtin_amdgcn_global_load_async_to_lds_b32(
        GAS32(y + (size_t)brow * DDIM + lane),
        LAS32(yrow + wave * DDIM + lane), 0, 0);
  }
  __builtin_amdgcn_s_wait_asynccnt(0);
#else
  if (lane < DDIM) yrow[wave * DDIM + lane] = y[(size_t)brow * DDIM + lane];
#endif
#if USE_TDM
  __builtin_amdgcn_s_wait_tensorcnt(0);
#endif
  __syncthreads();

  // ---- per-lane component: forward substitution in registers -------------
  // LDS word index = lane*153 + j; 153 coprime with 64 banks -> conflict-free.
  const float* c  = prow + wave * ROWF + lane * NPC;
  const float* yv = yrow + wave * DDIM;

  float z[DDIM];
  float hld2 = 0.0f, sumsq = 0.0f;   // half-log-det accumulated in log2
  int tri = 0;
  #pragma unroll
  for (int i = 0; i < DDIM; ++i) {
    const float* r = c + 1 + DDIM + tri;     // row i of packed lower tri
    float acc = yv[i] - c[1 + i];
    #pragma unroll
    for (int j = 0; j < i; ++j) acc -= r[j] * z[j];
    float m  = fmaxf(r[i], -15.0f);
    // softplus via raw v_exp/v_log (base 2): sp = ln2 * log2(1 + 2^(m*log2e))
    float sp = LN2 * FAST_LOG2(1.0f + FAST_EXP2(m * LOG2E));
    float zi = acc * __builtin_amdgcn_rcpf(sp);
    z[i] = zi;
    sumsq = __builtin_fmaf(zi, zi, sumsq);
    hld2 += FAST_LOG2(sp);
    tri += i + 1;
  }
  float clp = -0.5f * (sumsq + DDIM * LOG_2PI) - LN2 * hld2;  // component logp
  float pim = fmaxf(c[0], -15.0f);                            // clamped logit
  float s = pim + clp;

  // ---- out[b] = LSE_k(pim+clp) - LSE_k(pim) over the 32 lanes ------------
  float m1 = pim, m2 = s;
  #pragma unroll
  for (int off = 16; off > 0; off >>= 1) {
    m1 = fmaxf(m1, __shfl_xor(m1, off, 32));
    m2 = fmaxf(m2, __shfl_xor(m2, off, 32));
  }
  float e1 = FAST_EXP2((pim - m1) * LOG2E);
  float e2 = FAST_EXP2((s - m2) * LOG2E);

  // One v_wmma_f32_16x16x4_f32 reduces both 32-lane sums:
  //   A[m,{0,2}] = e2 (VGPR0), A[m,{1,3}] = e1 (VGPR1)
  //   B rows {0,2} = 1 on cols n<8 ; B rows {1,3} = 1 on cols n>=8
  //   => D[m,n<8] = e2_m + e2_{m+16},  D[m,n>=8] = e1_m + e1_{m+16}
  v2f a;  a[0] = e2; a[1] = e1;
  v2f bm;
  const int nl = lane & 15;
  bm[0] = (nl < 8) ? 1.0f : 0.0f;
  bm[1] = (nl < 8) ? 0.0f : 1.0f;
  v8f cacc = {};
  cacc = __builtin_amdgcn_wmma_f32_16x16x4_f32(false, a, false, bm,
                                               (short)0, cacc, false, false);
  float t = cacc[0] + cacc[1] + cacc[2] + cacc[3] +
            cacc[4] + cacc[5] + cacc[6] + cacc[7];   // rows 0-7 / 8-15
  t += __shfl_xor(t, 16, 32);    // lanes 0-7: sum(e2); lanes 8-15: sum(e1)
  float S2 = t;
  float S1 = __shfl(t, 8, 32);

  if (b < nrows && lane == 0)
    out[b] = (m2 + LN2 * FAST_LOG2(S2)) - (m1 + LN2 * FAST_LOG2(S1));
}

extern "C" void kernel_launch(void* const* d_in, const int* in_sizes, int n_in,
                              void* d_out, int out_size, void* d_ws, size_t ws_size,
                              hipStream_t stream) {
  const float* p = (const float*)d_in[0];   // (B, 4896) f32
  const float* y = (const float*)d_in[1];   // (B, 16)   f32
  float* out = (float*)d_out;               // (B,)      f32
  const int nrows = out_size;               // B = 16384

  const int blocks = (nrows + ROWS - 1) / ROWS;
  const size_t shmem = (size_t)(ROWS * ROWF + ROWS * DDIM) * sizeof(float);
  mdn_mv_logp_kernel<<<blocks, BLOCK, shmem, stream>>>(p, y, out, nrows);
}